// MultiBandMRI_32796370272263
// MI455X (gfx1250) — compile-verified
//
#include <hip/hip_runtime.h>
#include <math.h>

typedef float v2f __attribute__((ext_vector_type(2)));
typedef float v8f __attribute__((ext_vector_type(8)));

#define B_   2
#define C_   8
#define NB_  3
#define X_   80
#define Y_   80
#define M_   4096
#define XY_  (X_*Y_)
#define CB_  (C_*NB_)   /* 24 rows, padded to 32 (two 16-row WMMA tiles) */

#define WMMA4(A, Bm, Cacc) \
  __builtin_amdgcn_wmma_f32_16x16x4_f32(false, (A), false, (Bm), (short)0, (Cacc), false, false)

__global__ __launch_bounds__(32)
void mb_mri_fwd(const float* __restrict__ xin,    // [B,NB,X,Y,2]
                const float* __restrict__ maps,   // [B,C,NB,X,Y,2]
                const float* __restrict__ phi,    // [B,NB,M,2]
                const float* __restrict__ coord,  // [B,M,2]
                float* __restrict__ out)          // [B,C,M,2]
{
  // LDS staging: cimg chunk [cb(32)][y(80)+pad], Ey [n(16)][y(80)+pad],
  // Ex [x(80)][n(16)], and the final D tile for the phi reduction.
  __shared__ float2 ldsC [32*81];
  __shared__ float2 ldsEy[16*81];
  __shared__ float2 ldsEx[80*16];
  __shared__ float2 ldsD [32*16];

  const int lane = threadIdx.x;        // wave32, one wave per block
  const int mt   = blockIdx.x;         // 16-wide m tile
  const int b    = blockIdx.y;

  // ---- Build separable NUDFT factors: E[m, x*80+y] = Ex[m,x]*Ey[m,y] ----
  const float TWO_PI = 6.28318530717958647692f;
  for (int i = lane; i < 80*16; i += 32) {
    int q = i >> 4;                    // spatial index (x or y)
    int n = i & 15;                    // column within m-tile
    int m = mt*16 + n;
    float kx = coord[((size_t)b*M_ + m)*2 + 0];
    float ky = coord[((size_t)b*M_ + m)*2 + 1];
    float rr = (float)(q - 40);        // arange(80) - 80//2
    float sx, cx, sy, cy;
    __sincosf(-TWO_PI * kx * rr * (1.0f/X_), &sx, &cx);
    __sincosf(-TWO_PI * ky * rr * (1.0f/Y_), &sy, &cy);
    const float norm = 1.0f/80.0f;     // 1/sqrt(X*Y) folded into Ex
    ldsEx[q*16 + n] = make_float2(cx*norm, sx*norm);
    ldsEy[n*81 + q] = make_float2(cy,      sy);
  }
  // Zero-pad cimg rows 24..31 once (never rewritten).
  for (int i = lane; i < 8*80; i += 32)
    ldsC[(24 + i/80)*81 + (i - (i/80)*80)] = make_float2(0.f, 0.f);
  __syncthreads();

  // ---- Complex GEMM accumulators: D = cimg x E^T, split re/im parts ----
  v8f a_rr0 = {0,0,0,0,0,0,0,0}, a_ii0 = {0,0,0,0,0,0,0,0};
  v8f a_ri0 = {0,0,0,0,0,0,0,0}, a_ir0 = {0,0,0,0,0,0,0,0};
  v8f a_rr1 = {0,0,0,0,0,0,0,0}, a_ii1 = {0,0,0,0,0,0,0,0};
  v8f a_ri1 = {0,0,0,0,0,0,0,0}, a_ir1 = {0,0,0,0,0,0,0,0};

  const int n    = lane & 15;          // WMMA column (B/N) and row (A/M) for this lane
  const int ksel = (lane >> 4) << 1;   // f32 16x16x4 layout: lanes0-15 K={0,1}, lanes16-31 K={2,3}

  for (int x = 0; x < 80; ++x) {
    // Stage cimg chunk (maps*x, fixed image row x) into LDS, coalesced in y.
    for (int i = lane; i < CB_*80; i += 32) {
      int cb = i / 80, y = i - cb*80;
      int c  = cb / 3, nb = cb - c*3;
      int p  = x*80 + y;
      const float2 mv = *(const float2*)(maps + ((((size_t)b*C_ + c)*NB_ + nb)*XY_ + p)*2);
      const float2 xv = *(const float2*)(xin  + (((size_t)b*NB_ + nb)*XY_ + p)*2);
      ldsC[cb*81 + y] = make_float2(mv.x*xv.x - mv.y*xv.y,
                                    mv.x*xv.y + mv.y*xv.x);
    }
    if (x + 1 < 80 && lane < CB_) {    // prefetch next chunk (global_prefetch_b8)
      int c = lane / 3, nb = lane - c*3;
      __builtin_prefetch(maps + ((((size_t)b*C_ + c)*NB_ + nb)*XY_ + (x+1)*80)*2, 0, 1);
    }
    __syncthreads();

    const float2 exc = ldsEx[x*16 + n];     // constant over this chunk

    #pragma unroll 4
    for (int kk = 0; kk < 20; ++kk) {       // K = 80 per chunk, 4 per WMMA
      const int y0 = kk*4 + ksel;
      // B tile: E^T[k, m] = Ex[m,x]*Ey[m,y0+{0,1}]
      const float2 e0 = ldsEy[n*81 + y0];
      const float2 e1 = ldsEy[n*81 + y0 + 1];
      v2f Br, Bi;
      Br.x = exc.x*e0.x - exc.y*e0.y;  Bi.x = exc.x*e0.y + exc.y*e0.x;
      Br.y = exc.x*e1.x - exc.y*e1.y;  Bi.y = exc.x*e1.y + exc.y*e1.x;
      // A tiles: cimg rows n and 16+n at K = y0, y0+1
      const float2 a00 = ldsC[n*81 + y0];
      const float2 a01 = ldsC[n*81 + y0 + 1];
      const float2 a10 = ldsC[(16+n)*81 + y0];
      const float2 a11 = ldsC[(16+n)*81 + y0 + 1];
      v2f Ar0, Ai0, Ar1, Ai1;
      Ar0.x = a00.x; Ar0.y = a01.x;  Ai0.x = a00.y; Ai0.y = a01.y;
      Ar1.x = a10.x; Ar1.y = a11.x;  Ai1.x = a10.y; Ai1.y = a11.y;
      // Complex GEMM via 4 real WMMA per row-tile
      a_rr0 = WMMA4(Ar0, Br, a_rr0);
      a_ii0 = WMMA4(Ai0, Bi, a_ii0);
      a_ri0 = WMMA4(Ar0, Bi, a_ri0);
      a_ir0 = WMMA4(Ai0, Br, a_ir0);
      a_rr1 = WMMA4(Ar1, Br, a_rr1);
      a_ii1 = WMMA4(Ai1, Bi, a_ii1);
      a_ri1 = WMMA4(Ar1, Bi, a_ri1);
      a_ir1 = WMMA4(Ai1, Br, a_ir1);
    }
    __syncthreads();                   // before restaging ldsC
  }

  // ---- Combine complex parts and exchange through LDS ----
  v8f dre0 = a_rr0 - a_ii0, dim0 = a_ri0 + a_ir0;
  v8f dre1 = a_rr1 - a_ii1, dim1 = a_ri1 + a_ir1;
  const int colD = lane & 15;
  const int rh   = (lane >> 4) * 8;    // C/D layout: VGPR v, lanes16-31 => rows +8
  #pragma unroll
  for (int v = 0; v < 8; ++v) {
    ldsD[(rh + v)*16 + colD]      = make_float2(dre0[v], dim0[v]);
    ldsD[(16 + rh + v)*16 + colD] = make_float2(dre1[v], dim1[v]);
  }
  __syncthreads();

  // ---- Fused phi basis multiply + band sum: out[c,m] = sum_nb phi[nb,m]*D[c*3+nb,m]
  #pragma unroll
  for (int t = 0; t < 4; ++t) {
    int idx = lane + 32*t;             // 8 coils x 16 columns = 128 outputs
    int c   = idx >> 4, col = idx & 15;
    int m   = mt*16 + col;
    float orr = 0.f, oii = 0.f;
    #pragma unroll
    for (int nb = 0; nb < NB_; ++nb) {
      const float2 d = ldsD[(c*NB_ + nb)*16 + col];
      const float2 p = *(const float2*)(phi + (((size_t)b*NB_ + nb)*M_ + m)*2);
      orr += p.x*d.x - p.y*d.y;
      oii += p.x*d.y + p.y*d.x;
    }
    out[(((size_t)b*C_ + c)*M_ + m)*2 + 0] = orr;
    out[(((size_t)b*C_ + c)*M_ + m)*2 + 1] = oii;
  }
}

extern "C" void kernel_launch(void* const* d_in, const int* in_sizes, int n_in,
                              void* d_out, int out_size, void* d_ws, size_t ws_size,
                              hipStream_t stream) {
  (void)in_sizes; (void)n_in; (void)out_size; (void)d_ws; (void)ws_size;
  const float* xin   = (const float*)d_in[0];
  const float* maps  = (const float*)d_in[1];
  const float* phi   = (const float*)d_in[2];
  const float* coord = (const float*)d_in[3];
  float* out = (float*)d_out;
  dim3 grid(M_/16, B_);                // 256 m-tiles x 2 batches, 1 wave32 each
  mb_mri_fwd<<<grid, dim3(32,1,1), 0, stream>>>(xin, maps, phi, coord, out);
}